// GraphConv_9723805958477
// MI455X (gfx1250) — compile-verified
//
#include <hip/hip_runtime.h>

// Problem constants (from reference)
#define NODES 2000
#define EDGES 32000
#define ROWS  192            // B*T = 16*12
#define FIN   32
#define ROWF  (ROWS * FIN)   // 6144 floats per node block
#define FSTR  36             // padded LDS row stride (bank-conflict free, 16B aligned)

typedef __attribute__((ext_vector_type(2))) float v2f;
typedef __attribute__((ext_vector_type(4))) float v4f;
typedef __attribute__((ext_vector_type(8))) float v8f;

// ---------------- CSR construction ----------------
__global__ void k_zero(int* __restrict__ counts) {
    int i = blockIdx.x * blockDim.x + threadIdx.x;
    if (i < NODES) counts[i] = 0;
}

__global__ void k_count(const int* __restrict__ src, int* __restrict__ counts) {
    int e = blockIdx.x * blockDim.x + threadIdx.x;
    if (e < EDGES) atomicAdd(&counts[src[e]], 1);
}

__global__ void k_scan(const int* __restrict__ counts, int* __restrict__ offsets,
                       int* __restrict__ cursor) {
    if (threadIdx.x == 0 && blockIdx.x == 0) {
        int acc = 0;
        for (int i = 0; i < NODES; ++i) {
            offsets[i] = acc;
            cursor[i]  = acc;
            acc += counts[i];
        }
        offsets[NODES] = acc;
    }
}

__global__ void k_fill(const int* __restrict__ src, const int* __restrict__ dst,
                       int* __restrict__ cursor, int* __restrict__ csr) {
    int e = blockIdx.x * blockDim.x + threadIdx.x;
    if (e < EDGES) {
        int p = atomicAdd(&cursor[src[e]], 1);
        csr[p] = dst[e];
    }
}

// ---------------- fused gather-mean + dual GEMM + concat + ReLU ----------------
// One workgroup (8 wave32) per node.
__global__ __launch_bounds__(256) void k_conv(const float* __restrict__ x,
                                              const float* __restrict__ w,
                                              const int* __restrict__ counts,
                                              const int* __restrict__ offsets,
                                              const int* __restrict__ csr,
                                              float* __restrict__ out) {
    __shared__ float sFeat[2 * ROWS * FSTR]; // rows 0..191 = x[n], 192..383 = mean
    __shared__ float sWt[FIN * FSTR];        // weight transposed: sWt[col][k]

    const int node = blockIdx.x;
    const int t    = threadIdx.x;

    // Stage weight transposed (32x32)
    for (int i = t; i < FIN * FIN; i += 256) {
        int k = i >> 5, c = i & 31;
        sWt[c * FSTR + k] = w[i];
    }

    // Neighbor-sum accumulation, register resident: each thread owns 6 float4s
    v4f acc[6];
#pragma unroll
    for (int j = 0; j < 6; ++j) acc[j] = (v4f)0.0f;

    const int e0 = offsets[node];
    const int e1 = offsets[node + 1];
    for (int e = e0; e < e1; ++e) {
        const int d = csr[e]; // neighbor = x[edges_dst]
        const v4f* __restrict__ xd = (const v4f*)(x + (size_t)d * ROWF);
#pragma unroll
        for (int j = 0; j < 6; ++j) acc[j] += xd[t + j * 256];
    }

    const float recip = 1.0f / fmaxf((float)counts[node], 1.0f);
    const v4f* __restrict__ xs = (const v4f*)(x + (size_t)node * ROWF);
#pragma unroll
    for (int j = 0; j < 6; ++j) {
        const int idx = (t + j * 256) * 4;  // flat float index, multiple of 4
        const int row = idx >> 5;           // /FIN
        const int col = idx & 31;
        *(v4f*)&sFeat[row * FSTR + col]          = xs[t + j * 256]; // self features
        *(v4f*)&sFeat[(row + ROWS) * FSTR + col] = acc[j] * recip;  // mean features
    }
    __syncthreads();

    // GEMM: (384 x 32) @ (32 x 32) via V_WMMA_F32_16X16X4_F32
    // 24 M-tiles x 2 N-tiles = 48 tiles; 6 tiles per wave
    const int wave   = t >> 5;
    const int lane   = t & 31;
    const int laneLo = lane & 15;
    const int kOff   = (lane >> 4) << 1;   // +0 for lanes 0-15, +2 for lanes 16-31

    for (int tile = wave; tile < 48; tile += 8) {
        const int r0 = (tile >> 1) << 4;
        const int c0 = (tile & 1) << 4;

        const float* aBase = &sFeat[(r0 + laneLo) * FSTR];
        const float* bBase = &sWt[(c0 + laneLo) * FSTR];

        v8f cacc = {};
#pragma unroll
        for (int kk = 0; kk < 8; ++kk) {
            const int k0 = kk * 4 + kOff;
            v2f a = *(const v2f*)(aBase + k0); // A[M=lane&15][k0], [k0+1]
            v2f b = *(const v2f*)(bBase + k0); // B[k0][N=lane&15], B[k0+1][N]
            cacc = __builtin_amdgcn_wmma_f32_16x16x4_f32(
                false, a, false, b, (short)0, cacc, false, false);
        }

        // concat: self -> cols 0..31, aggregated -> cols 32..63; fused ReLU
        const int  orow0 = (r0 < ROWS) ? r0 : (r0 - ROWS);
        const int  ocol  = ((r0 < ROWS) ? 0 : 32) + c0 + laneLo;
        float* __restrict__ ob = out + (size_t)node * ROWS * 64;
        const int rhi = (lane >> 4) << 3;
#pragma unroll
        for (int j = 0; j < 8; ++j) {
            const int row = orow0 + j + rhi;
            ob[(size_t)row * 64 + ocol] = fmaxf(cacc[j], 0.0f);
        }
    }
}

// ---------------- launcher ----------------
extern "C" void kernel_launch(void* const* d_in, const int* in_sizes, int n_in,
                              void* d_out, int out_size, void* d_ws, size_t ws_size,
                              hipStream_t stream) {
    const float* x      = (const float*)d_in[0];
    const float* weight = (const float*)d_in[1];
    const int*   e_src  = (const int*)d_in[2];
    const int*   e_dst  = (const int*)d_in[3];
    float*       out    = (float*)d_out;

    // workspace layout (ints): counts[2000] | offsets[2001] | cursor[2000] | csr[32000]
    int* ws      = (int*)d_ws;
    int* counts  = ws;
    int* offsets = ws + NODES;
    int* cursor  = ws + NODES + NODES + 1;
    int* csr     = ws + NODES + NODES + 1 + NODES;

    k_zero<<<(NODES + 255) / 256, 256, 0, stream>>>(counts);
    k_count<<<(EDGES + 255) / 256, 256, 0, stream>>>(e_src, counts);
    k_scan<<<1, 1, 0, stream>>>(counts, offsets, cursor);
    k_fill<<<(EDGES + 255) / 256, 256, 0, stream>>>(e_src, e_dst, cursor, csr);
    k_conv<<<NODES, 256, 0, stream>>>(x, weight, counts, offsets, csr, out);
}